// Eikonal3D_15307263443051
// MI455X (gfx1250) — compile-verified
//
#include <hip/hip_runtime.h>
#include <cstdint>

// ---------------------------------------------------------------------------
// 3-D Eikonal (Godunov upwind, Jacobi sweeps) for MI455X / gfx1250.
// 96^3 f32 grid, 288 sweeps, source pinned at (48,48,48).
// Working set ~10 MB -> fully L2 resident; kernel is launch/L2-latency bound.
// CDNA5 path: Tensor Data Mover loads each block's 3-D halo tile into LDS
// (tensor_load_to_lds + s_wait_tensorcnt); f loads overlap the DMA; the
// 7-point stencil reads come from LDS.
// ---------------------------------------------------------------------------

#define NX      96
#define NXY     (NX * NX)
#define NTOT    (NX * NX * NX)
#define TX      16
#define TY      8
#define TZ      8
#define HALO_MAX ((TX + 2) * (TY + 2) * (TZ + 2))   // 18*10*10 = 1800 floats
#define BIGV    1.0e5f
#define N_ITER  288
#define SRC_IDX (48 + 48 * NX + 48 * NXY)

typedef __attribute__((ext_vector_type(4))) unsigned int tdm_v4u;
typedef __attribute__((ext_vector_type(8))) int          tdm_v8i;
typedef __attribute__((ext_vector_type(4))) int          tdm_v4i;

#if defined(__HIP_DEVICE_COMPILE__) && \
    __has_builtin(__builtin_amdgcn_tensor_load_to_lds) && \
    __has_builtin(__builtin_amdgcn_s_wait_tensorcnt)
#define EIK_USE_TDM 1
#else
#define EIK_USE_TDM 0
#endif

__global__ __launch_bounds__(256) void eik_step(const float* __restrict__ uin,
                                                const float* __restrict__ f,
                                                float* __restrict__ uout) {
  __shared__ float su[HALO_MAX];

  const int bx = blockIdx.x * TX, by = blockIdx.y * TY, bz = blockIdx.z * TZ;
  // Halo box clamped to the grid (TDM OOB reads return 0, but the algorithm
  // needs BIG at physical boundaries -> never let the TDM go out of bounds).
  const int x0 = (bx > 0) ? bx - 1 : 0;
  const int y0 = (by > 0) ? by - 1 : 0;
  const int z0 = (bz > 0) ? bz - 1 : 0;
  const int x1 = (bx + TX + 1 < NX) ? bx + TX + 1 : NX;
  const int y1 = (by + TY + 1 < NX) ? by + TY + 1 : NX;
  const int z1 = (bz + TZ + 1 < NX) ? bz + TZ + 1 : NX;
  const int ex = x1 - x0, ey = y1 - y0, ez = z1 - z0;   // packed LDS box dims
  const int ox = bx - x0, oy = by - y0, oz = bz - z0;   // interior offset in box

#if EIK_USE_TDM
  // Wave 0 issues a single 3-D tile DMA (global -> LDS) via the TDM and
  // waits on TENSORcnt; the other 7 waves run ahead to their f prefetches.
  if (threadIdx.x < 32) {
    const uint32_t lds_addr = (uint32_t)(uintptr_t)su;  // addrspace(3) byte offset
    const uint64_t gaddr = (uint64_t)(uintptr_t)uin +
                           4ull * (uint64_t)(x0 + NX * (y0 + NX * z0));
    // D# group 0: count=1(valid), lds_addr, 57-bit global tile addr, type=2.
    tdm_v4u g0 = { 1u,
                   lds_addr,
                   (uint32_t)gaddr,
                   (uint32_t)((gaddr >> 32) & 0x01FFFFFFu) | 0x80000000u };
    // D# group 1: wg_mask=0, data_size=2(4B), no pad/iterate/barrier,
    // tensor_dim0=96, tensor_dim1=96, tile dims ex/ey/ez,
    // dim0_stride=96, dim1_stride=96*96.
    tdm_v8i g1 = { (int)(2u << 16),                 // data_size = 4 bytes
                   (int)((NX & 0xFFFF) << 16),      // tensor_dim0[15:0]
                   (int)((NX & 0xFFFF) << 16),      // dim0[31:16]=0 | tensor_dim1[15:0]
                   (int)(ex << 16),                 // dim1[31:16]=0 | tile_dim0
                   (int)((ez << 16) | ey),          // tile_dim1 | tile_dim2
                   (int)NX,                         // tensor_dim0_stride[31:0]
                   (int)((NXY & 0xFFFF) << 16),     // d0s[47:32]=0 | dim1_stride[15:0]
                   (int)(NXY >> 16) };              // dim1_stride[47:16]
    // D# group 2 (3-D tensor): tensor_dim2=96, dim3=1, dim2_stride, tile_dim3=0.
    tdm_v4i g2 = { NX, 1, NTOT, 0 };
    tdm_v4i g3 = { 0, 0, 0, 0 };
    // Trailing int32x8 operand of the 6-arg (clang-23) builtin: zero-filled
    // (a zeroed descriptor group word = count=0 terminating NULL per ISA 8.3).
    tdm_v8i gz = { 0, 0, 0, 0, 0, 0, 0, 0 };
    __builtin_amdgcn_tensor_load_to_lds(g0, g1, g2, g3, gz, 0);
    __builtin_amdgcn_s_wait_tensorcnt(0);
  }
#else
  // Fallback: cooperative per-lane load of the packed halo box.
  const int nbox = ex * ey * ez;
  for (int i = threadIdx.x; i < nbox; i += 256) {
    int X = i % ex, t = i / ex;
    int Y = t % ey, Z = t / ey;
    su[i] = uin[(z0 + Z) * NXY + (y0 + Y) * NX + (x0 + X)];
  }
#endif

  // Prefetch f (independent of the LDS tile) to overlap with the TDM DMA.
  const int tid = threadIdx.x;
  int   gxa[4], gya[4], gza[4], gidxa[4];
  float fva[4];
#pragma unroll
  for (int k = 0; k < 4; ++k) {
    const int c  = tid + k * 256;          // 1024 cells per block
    const int lx = c & (TX - 1);
    const int ly = (c >> 4) & (TY - 1);
    const int lz = c >> 7;
    gxa[k] = bx + lx;  gya[k] = by + ly;  gza[k] = bz + lz;
    gidxa[k] = gxa[k] + gya[k] * NX + gza[k] * NXY;
    fva[k] = f[gidxa[k]];                  // H == 1.0, so fh == f
  }

  __syncthreads();

  const int exy = ex * ey;

#pragma unroll
  for (int k = 0; k < 4; ++k) {
    const int gx = gxa[k], gy = gya[k], gz = gza[k];
    const int lx = gx - bx, ly = gy - by, lz = gz - bz;

    const int sidx = ((lz + oz) * ey + (ly + oy)) * ex + (lx + ox);
    const float uc = su[sidx];
    const float xm = (gx > 0)      ? su[sidx - 1]   : BIGV;
    const float xp = (gx < NX - 1) ? su[sidx + 1]   : BIGV;
    const float ym = (gy > 0)      ? su[sidx - ex]  : BIGV;
    const float yp = (gy < NX - 1) ? su[sidx + ex]  : BIGV;
    const float zm = (gz > 0)      ? su[sidx - exy] : BIGV;
    const float zp = (gz < NX - 1) ? su[sidx + exy] : BIGV;

    const float a  = fminf(xm, xp);
    const float b  = fminf(ym, yp);
    const float cc = fminf(zm, zp);
    const float lo  = fminf(fminf(a, b), cc);
    const float hi  = fmaxf(fmaxf(a, b), cc);
    const float mid = a + b + cc - lo - hi;

    const float fh = fva[k];

    const float u1  = lo + fh;
    const float dlm = lo - mid;
    const float d2  = 2.0f * fh * fh - dlm * dlm;
    const float u2  = (d2 > 0.0f) ? 0.5f * (lo + mid + __builtin_sqrtf(d2)) : BIGV;
    const float s   = lo + mid + hi;
    const float d3  = s * s - 3.0f * (lo * lo + mid * mid + hi * hi - fh * fh);
    const float u3  = (d3 > 0.0f) ? (s + __builtin_sqrtf(d3)) / 3.0f : BIGV;

    const float cand = (u1 <= mid) ? u1 : ((u2 <= hi) ? u2 : u3);
    float un = fminf(uc, cand);
    if (gidxa[k] == SRC_IDX) un = 0.0f;               // pin source every sweep
    uout[gidxa[k]] = un;
  }
}

__global__ __launch_bounds__(256) void eik_init(float* __restrict__ u) {
  int i = blockIdx.x * blockDim.x + threadIdx.x;
  if (i < NTOT) u[i] = (i == SRC_IDX) ? 0.0f : BIGV;
}

extern "C" void kernel_launch(void* const* d_in, const int* in_sizes, int n_in,
                              void* d_out, int out_size, void* d_ws, size_t ws_size,
                              hipStream_t stream) {
  (void)in_sizes; (void)n_in; (void)out_size; (void)ws_size;

  const float* f   = (const float*)d_in[0];
  float*       out = (float*)d_out;
  float*       ua  = (float*)d_ws;           // ping
  float*       ub  = ua + NTOT;              // pong  (needs 2*3.54 MB of d_ws)

  eik_init<<<(NTOT + 255) / 256, 256, 0, stream>>>(ua);

  const dim3 grid(NX / TX, NX / TY, NX / TZ);   // 6 x 12 x 12 = 864 blocks
  float* bufs[2] = { ua, ub };
  const float* cur = ua;
  for (int it = 0; it < N_ITER; ++it) {
    float* nxt = (it == N_ITER - 1) ? out : bufs[(it + 1) & 1];
    eik_step<<<grid, 256, 0, stream>>>(cur, f, nxt);
    cur = nxt;
  }
}